// HeteroGNN_43490838839759
// MI455X (gfx1250) — compile-verified
//
#include <hip/hip_runtime.h>
#include <hip/hip_bf16.h>

// ---------------------------------------------------------------------------
// HeteroGNN forward, MI455X (gfx1250, wave32).
// Dense 128-wide GEMMs run on V_WMMA_F32_16X16X4_F32 (full f32 precision,
// matrix pipe). Edge scatter/softmax are bandwidth-bound wave-per-edge
// kernels with f32 global atomics. All launches on `stream`; scratch in d_ws.
// ---------------------------------------------------------------------------

typedef float v2f __attribute__((ext_vector_type(2)));
typedef float v8f __attribute__((ext_vector_type(8)));

#define H128 128
#define CN1 40000
#define CN2 40000
#define CE  640000
#define CG  64

#define MODE_BIAS 0
#define MODE_BN_RELU 1
#define MODE_RELU 2
#define MODE_ADD_SELF_RELU 3

// ---------------- WMMA GEMM: Y[M,128] = X[M,K] @ W[K,128] (+epilogue) -------
// 256 threads = 8 waves; wave w owns cols [16w,16w+16); block owns 16 rows.
__global__ __launch_bounds__(256)
void k_gemm_wmma(const float* __restrict__ X, const float* __restrict__ W,
                 const float* __restrict__ bias, float* __restrict__ Y,
                 int M, int K, int mode,
                 const float* __restrict__ bn_g, const float* __restrict__ bn_b,
                 const float* __restrict__ bn_rm, const float* __restrict__ bn_rv)
{
    const int N = H128;
    const int lane = threadIdx.x & 31;
    const int wv   = threadIdx.x >> 5;
    const int m0   = blockIdx.x * 16;
    if (m0 >= M) return;
    const int col0 = wv * 16;
    const int half = lane >> 4;      // 0: K offsets 0,1 ; 1: K offsets 2,3
    const int l16  = lane & 15;
    const int col  = col0 + l16;

    const float* xrow = X + (size_t)(m0 + l16) * K;   // A: row m0+l16
    v8f acc = {};
    for (int k = 0; k < K; k += 4) {
        v2f a, b;
        const float* ap = xrow + k + 2 * half;
        a.x = ap[0];  a.y = ap[1];                    // A 16x4 fragment
        const float* bp = W + (size_t)(k + 2 * half) * N + col;
        b.x = bp[0];  b.y = bp[N];                    // B 4x16 fragment
        acc = __builtin_amdgcn_wmma_f32_16x16x4_f32(
                  false, a, false, b, (short)0, acc, false, false);
    }

    // Hoist per-column epilogue constants.
    float bval = bias ? bias[col] : 0.0f;
    float scale = 1.0f, shift = 0.0f;
    if (mode == MODE_BN_RELU) {
        scale = bn_g[col] * rsqrtf(bn_rv[col] + 1e-5f);
        shift = bn_b[col] - bn_rm[col] * scale;
    }
    #pragma unroll
    for (int i = 0; i < 8; ++i) {
        int row = m0 + i + 8 * half;                  // C/D layout (ISA 7.12.2)
        float v = acc[i] + bval;
        if (mode == MODE_BN_RELU)            v = fmaxf(v * scale + shift, 0.0f);
        else if (mode == MODE_RELU)          v = fmaxf(v, 0.0f);
        else if (mode == MODE_ADD_SELF_RELU) v = fmaxf(v + X[(size_t)row * K + col], 0.0f);
        Y[(size_t)row * N + col] = v;
    }
}

// ---------------- utility / elementwise ------------------------------------
__global__ void k_fill(float* __restrict__ p, float v, long long n) {
    long long i = (long long)blockIdx.x * blockDim.x + threadIdx.x;
    long long s = (long long)gridDim.x * blockDim.x;
    for (; i < n; i += s) p[i] = v;
}

__global__ void k_deg(const int* __restrict__ dst, float* __restrict__ deg, int e) {
    int i = blockIdx.x * blockDim.x + threadIdx.x;
    if (i < e) atomicAdd(&deg[dst[i]], 1.0f);
}

__global__ void k_dinv(float* __restrict__ d, int n) {
    int i = blockIdx.x * blockDim.x + threadIdx.x;
    if (i < n) d[i] = rsqrtf(d[i] + 1.0f);
}

// GCN: n[dst] += dinv[s]*dinv[d] * hW[src]  (wave per edge, float4 per lane)
__global__ void k_gcn_scatter(const int* __restrict__ src, const int* __restrict__ dst,
                              const float* __restrict__ dinv, const float* __restrict__ hw,
                              float* __restrict__ outn, int e_count) {
    int tid = blockIdx.x * blockDim.x + threadIdx.x;
    int e = tid >> 5, lane = tid & 31;
    if (e >= e_count) return;
    int s = src[e], d = dst[e];
    float norm = dinv[s] * dinv[d];
    const float* hp = hw   + (size_t)s * H128 + lane * 4;
    float*       op = outn + (size_t)d * H128 + lane * 4;
    #pragma unroll
    for (int j = 0; j < 4; ++j) atomicAdd(op + j, norm * hp[j]);
}

// n += dinv^2 * hW + b_gcn + b_gat   (self-loop term + both edge-type biases)
__global__ void k_self_bias(float* __restrict__ n, const float* __restrict__ hw,
                            const float* __restrict__ dinv,
                            const float* __restrict__ b1, const float* __restrict__ b2,
                            int nodes) {
    int idx = blockIdx.x * blockDim.x + threadIdx.x;
    if (idx >= nodes * H128) return;
    int i = idx >> 7, c = idx & 127;
    float di = dinv[i];
    n[idx] += di * di * hw[idx] + b1[c] + b2[c];
}

// out[node] = dot(Hm[node,:128], a)   (wave per node)
__global__ void k_matvec(const float* __restrict__ Hm, const float* __restrict__ a,
                         float* __restrict__ out, int n) {
    int tid = blockIdx.x * blockDim.x + threadIdx.x;
    int node = tid >> 5, lane = tid & 31;
    if (node >= n) return;
    const float* row = Hm + (size_t)node * H128;
    float p = 0.0f;
    #pragma unroll
    for (int j = 0; j < 4; ++j) p += row[lane + 32 * j] * a[lane + 32 * j];
    #pragma unroll
    for (int off = 16; off > 0; off >>= 1) p += __shfl_xor(p, off, 32);
    if (lane == 0) out[node] = p;
}

__device__ __forceinline__ void atomicMaxF(float* addr, float val) {
    if (val >= 0.0f) atomicMax((int*)addr, __float_as_int(val));
    else             atomicMin((unsigned int*)addr, __float_as_uint(val));
}

__global__ void k_att_logit(const int* __restrict__ src, const int* __restrict__ dst,
                            const float* __restrict__ ls, const float* __restrict__ ld,
                            float* __restrict__ elog, float* __restrict__ m, int e) {
    int i = blockIdx.x * blockDim.x + threadIdx.x;
    if (i >= e) return;
    float l = ls[src[i]] + ld[dst[i]];
    l = (l > 0.0f) ? l : 0.2f * l;          // leaky_relu 0.2
    elog[i] = l;
    atomicMaxF(&m[dst[i]], l);
}

__global__ void k_fix_m(float* __restrict__ m, int n) {
    int i = blockIdx.x * blockDim.x + threadIdx.x;
    if (i < n && m[i] <= -3.3e38f) m[i] = 0.0f;   // empty segment -> 0
}

__global__ void k_att_exp(const int* __restrict__ dst, float* __restrict__ elog,
                          const float* __restrict__ m, float* __restrict__ den, int e) {
    int i = blockIdx.x * blockDim.x + threadIdx.x;
    if (i >= e) return;
    int d = dst[i];
    float ex = expf(elog[i] - m[d]);
    elog[i] = ex;                            // reuse scratch in place
    atomicAdd(&den[d], ex);
}

// n[dst] += alpha * hs[src]   (wave per edge)
__global__ void k_att_scatter(const int* __restrict__ src, const int* __restrict__ dst,
                              const float* __restrict__ ex, const float* __restrict__ den,
                              const float* __restrict__ hs, float* __restrict__ outn,
                              int e_count) {
    int tid = blockIdx.x * blockDim.x + threadIdx.x;
    int e = tid >> 5, lane = tid & 31;
    if (e >= e_count) return;
    int s = src[e], d = dst[e];
    float alpha = ex[e] / (den[d] + 1e-16f);
    const float* hp = hs   + (size_t)s * H128 + lane * 4;
    float*       op = outn + (size_t)d * H128 + lane * 4;
    #pragma unroll
    for (int j = 0; j < 4; ++j) atomicAdd(op + j, alpha * hp[j]);
}

// Mean-pool scatter: pool[batch[i]] += h[i]; cnt[batch[i]] += 1
__global__ void k_pool_scatter(const float* __restrict__ h, const int* __restrict__ batch,
                               float* __restrict__ pool, float* __restrict__ cnt, int n) {
    int tid = blockIdx.x * blockDim.x + threadIdx.x;
    int node = tid >> 5, lane = tid & 31;
    if (node >= n) return;
    int g = batch[node];
    const float* hp = h + (size_t)node * H128 + lane * 4;
    float* pp = pool + (size_t)g * H128 + lane * 4;
    #pragma unroll
    for (int j = 0; j < 4; ++j) atomicAdd(pp + j, hp[j]);
    if (lane == 0) atomicAdd(&cnt[g], 1.0f);
}

// pooled[g, 0:128]=pool1/c1 ; pooled[g,128:256]=pool2/c2
__global__ void k_pool_finish(const float* __restrict__ p1, const float* __restrict__ p2,
                              const float* __restrict__ c1, const float* __restrict__ c2,
                              float* __restrict__ pooled) {
    int idx = blockIdx.x * blockDim.x + threadIdx.x;  // 64*256
    int g = idx >> 8, c = idx & 255;
    float v = (c < 128) ? p1[g * 128 + c] / fmaxf(c1[g], 1.0f)
                        : p2[g * 128 + (c - 128)] / fmaxf(c2[g], 1.0f);
    pooled[g * 256 + c] = v;
}

// out[64,2] = z[64,128] @ W2[128,2] + b2
__global__ void k_lin2(const float* __restrict__ z, const float* __restrict__ W,
                       const float* __restrict__ b, float* __restrict__ out) {
    int t = threadIdx.x;         // 128 threads
    int g = t >> 1, o = t & 1;
    float s = b[o];
    const float* zr = z + g * 128;
    for (int c = 0; c < 128; ++c) s += zr[c] * W[c * 2 + o];
    out[g * 2 + o] = s;
}

// ---------------------------------------------------------------------------
extern "C" void kernel_launch(void* const* d_in, const int* in_sizes, int n_in,
                              void* d_out, int out_size, void* d_ws, size_t ws_size,
                              hipStream_t stream) {
    const float* x1 = (const float*)d_in[0];
    const float* x2 = (const float*)d_in[1];
    const int* ei11 = (const int*)d_in[2];
    const int* ei22 = (const int*)d_in[3];
    const int* ei12 = (const int*)d_in[4];
    const int* ei21 = (const int*)d_in[5];
    const int* batch1 = (const int*)d_in[6];
    const int* batch2 = (const int*)d_in[7];
    auto P = [&](int i) { return (const float*)d_in[i]; };

    // Param leaves: jax-style flatten (dict keys sorted, lists in order), base=8.
    // bn_post.g1:[8..11](b,g,rm,rv) g2:[12..15] | bn_pre.g1:[16..19] g2:[20..23]
    // convs[l] @ 24+14l: gat12(Wd,Ws,a_d,a_s,b) gat21(Wd,Ws,a_d,a_s,b)
    //                    gcn11(W,b) gcn22(W,b)
    // lin1:[66,67] lin2:[68,69] post.g1:[70,71] g2:[72,73]
    // pre.g1:[74,75] g2:[76,77] | skip[l] @ 78+4l: g1.W,g1.b,g2.W,g2.b

    // ---- workspace layout (floats) ----
    float* ws = (float*)d_ws;
    const size_t NB = (size_t)CN1 * H128;          // 5.12M floats
    float* h1 = ws;            float* h2 = ws + NB;
    float* n1 = ws + 2 * NB;   float* n2 = ws + 3 * NB;
    float* t0 = ws + 4 * NB;   float* t1 = ws + 5 * NB;
    float* eS    = ws + 6 * NB;          // [E] logits -> exp
    float* lsb   = eS + CE;              // [40000]
    float* ldb   = lsb + CN1;
    float* mbuf  = ldb + CN1;
    float* den   = mbuf + CN1;
    float* dinv1 = den + CN1;
    float* dinv2 = dinv1 + CN1;
    float* pool1 = dinv2 + CN2;          // 64*128
    float* pool2 = pool1 + CG * H128;
    float* cnt1  = pool2 + CG * H128;    // 64
    float* cnt2  = cnt1 + CG;
    float* pooled = cnt2 + CG;           // 64*256
    float* zbuf   = pooled + CG * 256;   // 64*128

    const dim3 B256(256);
    const int GB_E_T   = (CE + 255) / 256;        // thread-per-edge
    const int GB_E_W   = (CE * 32 + 255) / 256;   // wave-per-edge
    const int GB_N_W   = (CN1 * 32 + 255) / 256;  // wave-per-node
    const int GB_N_EL  = (CN1 * H128 + 255) / 256;
    const int GB_N_T   = (CN1 + 255) / 256;

    auto FILL = [&](float* p, float v, long long n) {
        k_fill<<<1024, B256, 0, stream>>>(p, v, n);
    };
    auto GEMM = [&](const float* X, const float* W, const float* b, float* Y,
                    int M, int K, int mode,
                    const float* g = nullptr, const float* bb = nullptr,
                    const float* rm = nullptr, const float* rv = nullptr) {
        k_gemm_wmma<<<dim3((M + 15) / 16), B256, 0, stream>>>(
            X, W, b, Y, M, K, mode, g, bb, rm, rv);
    };

    // ---- degrees (GCN symmetric norm), once ----
    FILL(dinv1, 0.0f, CN1); FILL(dinv2, 0.0f, CN2);
    k_deg<<<GB_E_T, B256, 0, stream>>>(ei11 + CE, dinv1, CE);
    k_deg<<<GB_E_T, B256, 0, stream>>>(ei22 + CE, dinv2, CE);
    k_dinv<<<GB_N_T, B256, 0, stream>>>(dinv1, CN1);
    k_dinv<<<GB_N_T, B256, 0, stream>>>(dinv2, CN2);

    // ---- pre linear + BN(eval) + relu ----
    GEMM(x1, P(74), P(75), h1, CN1, 64, MODE_BN_RELU, P(17), P(16), P(18), P(19));
    GEMM(x2, P(76), P(77), h2, CN2, 64, MODE_BN_RELU, P(21), P(20), P(22), P(23));

    for (int l = 0; l < 3; ++l) {
        int cb = 24 + 14 * l, sb = 78 + 4 * l;
        const float *g12_Wd = P(cb + 0), *g12_Ws = P(cb + 1), *g12_ad = P(cb + 2),
                    *g12_as = P(cb + 3), *g12_b = P(cb + 4);
        const float *g21_Wd = P(cb + 5), *g21_Ws = P(cb + 6), *g21_ad = P(cb + 7),
                    *g21_as = P(cb + 8), *g21_b = P(cb + 9);
        const float *gcn11_W = P(cb + 10), *gcn11_b = P(cb + 11);
        const float *gcn22_W = P(cb + 12), *gcn22_b = P(cb + 13);

        FILL(n1, 0.0f, (long long)NB);
        FILL(n2, 0.0f, (long long)NB);

        // ---- dst type 1: GCN(ei11) + GAT21(h2 -> h1) ----
        GEMM(h1, gcn11_W, nullptr, t0, CN1, 128, MODE_BIAS);
        k_gcn_scatter<<<GB_E_W, B256, 0, stream>>>(ei11, ei11 + CE, dinv1, t0, n1, CE);
        k_self_bias<<<GB_N_EL, B256, 0, stream>>>(n1, t0, dinv1, gcn11_b, g21_b, CN1);

        GEMM(h2, g21_Ws, nullptr, t0, CN2, 128, MODE_BIAS);   // hs (src=type2)
        GEMM(h1, g21_Wd, nullptr, t1, CN1, 128, MODE_BIAS);   // hd (dst=type1)
        k_matvec<<<GB_N_W, B256, 0, stream>>>(t0, g21_as, lsb, CN2);
        k_matvec<<<GB_N_W, B256, 0, stream>>>(t1, g21_ad, ldb, CN1);
        FILL(mbuf, -3.402823466e38f, CN1); FILL(den, 0.0f, CN1);
        k_att_logit<<<GB_E_T, B256, 0, stream>>>(ei21, ei21 + CE, lsb, ldb, eS, mbuf, CE);
        k_fix_m<<<GB_N_T, B256, 0, stream>>>(mbuf, CN1);
        k_att_exp<<<GB_E_T, B256, 0, stream>>>(ei21 + CE, eS, mbuf, den, CE);
        k_att_scatter<<<GB_E_W, B256, 0, stream>>>(ei21, ei21 + CE, eS, den, t0, n1, CE);

        // ---- dst type 2: GCN(ei22) + GAT12(h1 -> h2) ----
        GEMM(h2, gcn22_W, nullptr, t0, CN2, 128, MODE_BIAS);
        k_gcn_scatter<<<GB_E_W, B256, 0, stream>>>(ei22, ei22 + CE, dinv2, t0, n2, CE);
        k_self_bias<<<GB_N_EL, B256, 0, stream>>>(n2, t0, dinv2, gcn22_b, g12_b, CN2);

        GEMM(h1, g12_Ws, nullptr, t0, CN1, 128, MODE_BIAS);   // hs (src=type1)
        GEMM(h2, g12_Wd, nullptr, t1, CN2, 128, MODE_BIAS);   // hd (dst=type2)
        k_matvec<<<GB_N_W, B256, 0, stream>>>(t0, g12_as, lsb, CN1);
        k_matvec<<<GB_N_W, B256, 0, stream>>>(t1, g12_ad, ldb, CN2);
        FILL(mbuf, -3.402823466e38f, CN2); FILL(den, 0.0f, CN2);
        k_att_logit<<<GB_E_T, B256, 0, stream>>>(ei12, ei12 + CE, lsb, ldb, eS, mbuf, CE);
        k_fix_m<<<GB_N_T, B256, 0, stream>>>(mbuf, CN2);
        k_att_exp<<<GB_E_T, B256, 0, stream>>>(ei12 + CE, eS, mbuf, den, CE);
        k_att_scatter<<<GB_E_W, B256, 0, stream>>>(ei12, ei12 + CE, eS, den, t0, n2, CE);

        // ---- skip: h = relu(n + n@Wskip + b) ----
        GEMM(n1, P(sb + 0), P(sb + 1), h1, CN1, 128, MODE_ADD_SELF_RELU);
        GEMM(n2, P(sb + 2), P(sb + 3), h2, CN2, 128, MODE_ADD_SELF_RELU);
    }

    // ---- post linear + BN + relu (into n1/n2) ----
    GEMM(h1, P(70), P(71), n1, CN1, 128, MODE_BN_RELU, P(9),  P(8),  P(10), P(11));
    GEMM(h2, P(72), P(73), n2, CN2, 128, MODE_BN_RELU, P(13), P(12), P(14), P(15));

    // ---- mean pool per graph + concat ----
    FILL(pool1, 0.0f, CG * H128); FILL(pool2, 0.0f, CG * H128);
    FILL(cnt1, 0.0f, CG); FILL(cnt2, 0.0f, CG);
    k_pool_scatter<<<GB_N_W, B256, 0, stream>>>(n1, batch1, pool1, cnt1, CN1);
    k_pool_scatter<<<GB_N_W, B256, 0, stream>>>(n2, batch2, pool2, cnt2, CN2);
    k_pool_finish<<<64, B256, 0, stream>>>(pool1, pool2, cnt1, cnt2, pooled);

    // ---- head: lin1 (256->128, relu) on WMMA, then tiny lin2 (128->2) ----
    GEMM(pooled, P(66), P(67), zbuf, CG, 256, MODE_RELU);
    k_lin2<<<1, 128, 0, stream>>>(zbuf, P(68), P(69), (float*)d_out);
}